// ECVQ_17884243821128
// MI455X (gfx1250) — compile-verified
//
#include <hip/hip_runtime.h>
#include <math.h>

#define CB 6
#define CB_SIZE 1024
#define CB_DIM 8
#define LMBDA 0.005f
#define INV_LOG2 1.44269504088896340736f
#define FLT_BIG 3.402823466e+38f

typedef __attribute__((ext_vector_type(2))) float v2f;
typedef __attribute__((ext_vector_type(4))) float v4f;
typedef __attribute__((ext_vector_type(8))) float v8f;

// ---------------------------------------------------------------------------
// Prep: per-codebook column constants, packed as float2 per codeword:
//   colc[k,s].x = ||codebook[k,s]||^2
//   colc[k,s].y = -log_softmax(logits[k])[s] / ln2 / lambda  (added AFTER sqrt)
// ---------------------------------------------------------------------------
__global__ void __launch_bounds__(256) ecvq_prep(
    const float* __restrict__ codebook, const float* __restrict__ logits,
    v2f* __restrict__ colc) {
  const int k = blockIdx.x;
  const int t = threadIdx.x;
  __shared__ float red[256];

  float mx = -INFINITY;
  for (int s = t; s < CB_SIZE; s += 256) mx = fmaxf(mx, logits[k * CB_SIZE + s]);
  red[t] = mx; __syncthreads();
  for (int o = 128; o > 0; o >>= 1) {
    if (t < o) red[t] = fmaxf(red[t], red[t + o]);
    __syncthreads();
  }
  mx = red[0]; __syncthreads();

  float se = 0.f;
  for (int s = t; s < CB_SIZE; s += 256) se += __expf(logits[k * CB_SIZE + s] - mx);
  red[t] = se; __syncthreads();
  for (int o = 128; o > 0; o >>= 1) {
    if (t < o) red[t] += red[t + o];
    __syncthreads();
  }
  const float lse = mx + __logf(red[0]);

  for (int s = t; s < CB_SIZE; s += 256) {
    const float* c = codebook + (size_t)(k * CB_SIZE + s) * CB_DIM;
    float acc = 0.f;
#pragma unroll
    for (int d = 0; d < CB_DIM; ++d) acc += c[d] * c[d];
    v2f cc;
    cc.x = acc;
    cc.y = (lse - logits[k * CB_SIZE + s]) * INV_LOG2 * (1.0f / LMBDA);
    colc[k * CB_SIZE + s] = cc;
  }
}

// ---------------------------------------------------------------------------
// Main: one wave per (16-row tile, codebook). 64 column tiles, 2 WMMAs each.
// ---------------------------------------------------------------------------
__global__ void __launch_bounds__(256) ecvq_main(
    const float* __restrict__ x, const float* __restrict__ codebook,
    const v2f* __restrict__ colc,
    float* __restrict__ xhat_out, float* __restrict__ idx_out,
    float* __restrict__ prob_ws, int totalWaves) {
  const int lane = threadIdx.x & 31;
  const int wave = threadIdx.x >> 5;
  const int gw = blockIdx.x * 8 + wave;
  if (gw >= totalWaves) return;  // uniform per wave: EXEC stays all-ones

  const int rowTile = gw / CB;
  const int cb = gw - rowTile * CB;
  const int rowBase = rowTile * 16;
  const int hi = lane >> 4;   // which 16-lane half
  const int lj = lane & 15;

  // A fragment: 16x4 f32 layout -> lane holds row rowBase+lj, K={2hi,2hi+1}
  const float* xrow = x + (size_t)(rowBase + lj) * (CB * CB_DIM) + cb * CB_DIM;
  v2f a1, a2;
  a1.x = xrow[2 * hi + 0]; a1.y = xrow[2 * hi + 1];         // d in [0,4)
  a2.x = xrow[4 + 2 * hi + 0]; a2.y = xrow[4 + 2 * hi + 1]; // d in [4,8)

  // ||x||^2 for row rowBase+lj: combine this lane's 4 squares with mirror lane
  float part = a1.x * a1.x + a1.y * a1.y + a2.x * a2.x + a2.y * a2.y;
  float x2full = part + __shfl_xor(part, 16, 32);
  // re-index from N-layout (lane lj) to C-matrix M-layout (VGPR r, half hi)
  float x2v[8];
#pragma unroll
  for (int r = 0; r < 8; ++r) x2v[r] = __shfl(x2full, r + 8 * hi, 32);

  // fold the -2 of ||x-c||^2 into A so WMMA yields -2*x.c directly
  a1 *= -2.0f; a2 *= -2.0f;

  const float* cbk = codebook + (size_t)cb * CB_SIZE * CB_DIM;
  const v2f* cck = colc + cb * CB_SIZE;

  float bestCost[8]; int bestIdx[8];
#pragma unroll
  for (int r = 0; r < 8; ++r) { bestCost[r] = INFINITY; bestIdx[r] = 0x7FFFFFFF; }

  for (int ct = 0; ct < CB_SIZE / 16; ++ct) {
    const int myCol = ct * 16 + lj;
    const float* crow = cbk + (size_t)myCol * CB_DIM;
    v2f b1, b2;
    b1.x = crow[2 * hi + 0]; b1.y = crow[2 * hi + 1];
    b2.x = crow[4 + 2 * hi + 0]; b2.y = crow[4 + 2 * hi + 1];
    const v2f cc = cck[myCol];        // {||c||^2, penalty}
    const float c2l = cc.x;
    const float penl = cc.y;

    // Seed the accumulator with the bias (||x||^2 + ||c||^2) so the WMMA
    // chain directly produces q = ||x||^2 - 2 x.c + ||c||^2.
    v8f acc;
#pragma unroll
    for (int r = 0; r < 8; ++r) acc[r] = x2v[r] + c2l;
    acc = __builtin_amdgcn_wmma_f32_16x16x4_f32(
        false, a1, false, b1, (short)0, acc, false, false);
    acc = __builtin_amdgcn_wmma_f32_16x16x4_f32(
        false, a2, false, b2, (short)0, acc, false, false);

#pragma unroll
    for (int r = 0; r < 8; ++r) {
      // single-instruction clamp to [0, FLT_MAX] (v_med3_f32), then one
      // v_sqrt_f32 (~1 ulp) — distance comparison, not IEEE-critical
      const float q = __builtin_amdgcn_fmed3f(acc[r], 0.0f, FLT_BIG);
      const float cost = __builtin_amdgcn_sqrtf(q) + penl;
      if (cost < bestCost[r]) { bestCost[r] = cost; bestIdx[r] = myCol; }
    }
  }

  // argmin across the 16 lanes (columns) of each half; first-index tie-break
#pragma unroll
  for (int r = 0; r < 8; ++r) {
    float c = bestCost[r]; int i = bestIdx[r];
#pragma unroll
    for (int m = 8; m >= 1; m >>= 1) {
      const float oc = __shfl_xor(c, m, 32);
      const int   oi = __shfl_xor(i, m, 32);
      if (oc < c || (oc == c && oi < i)) { c = oc; i = oi; }
    }
    bestCost[r] = c; bestIdx[r] = i;
  }

  // writer lane lj==r of each half emits row rowBase + r + 8*hi
#pragma unroll
  for (int r = 0; r < 8; ++r) {
    if (lj == r) {
      const int row = rowBase + r + 8 * hi;
      const int bi = bestIdx[r];
      const float* src = cbk + (size_t)bi * CB_DIM;
      float* dst = xhat_out + (size_t)row * (CB * CB_DIM) + cb * CB_DIM;
      *(v4f*)dst       = *(const v4f*)src;
      *(v4f*)(dst + 4) = *(const v4f*)(src + 4);
      idx_out[(size_t)row * CB + cb] = (float)bi;
      prob_ws[(size_t)row * CB + cb] = cck[bi].y * LMBDA;  // -log2_pmf (bits) at argmin
    }
  }
}

// ---------------------------------------------------------------------------
// Deterministic single-block sum of per-(b,cb) bit costs -> scalar output.
// ---------------------------------------------------------------------------
__global__ void __launch_bounds__(256) ecvq_bits(
    const float* __restrict__ prob_ws, float* __restrict__ bits_out, int n) {
  __shared__ float red[256];
  float s = 0.f;
  for (int i = threadIdx.x; i < n; i += 256) s += prob_ws[i];
  red[threadIdx.x] = s; __syncthreads();
  for (int o = 128; o > 0; o >>= 1) {
    if (threadIdx.x < o) red[threadIdx.x] += red[threadIdx.x + o];
    __syncthreads();
  }
  if (threadIdx.x == 0) bits_out[0] = red[0];
}

extern "C" void kernel_launch(void* const* d_in, const int* in_sizes, int n_in,
                              void* d_out, int out_size, void* d_ws, size_t ws_size,
                              hipStream_t stream) {
  const float* x        = (const float*)d_in[0];
  const float* codebook = (const float*)d_in[1];
  const float* logits   = (const float*)d_in[2];
  const int b = in_sizes[0] / (CB * CB_DIM);

  // workspace layout
  v2f*   colc = (v2f*)d_ws;                         // CB*CB_SIZE float2
  float* prob = (float*)(colc + CB * CB_SIZE);      // b*CB floats

  // output layout: x_hat (b*48) | bits (1) | index-as-float (b*6)
  float* out  = (float*)d_out;
  float* xhat = out;
  float* bits = out + (size_t)b * CB * CB_DIM;
  float* idxo = bits + 1;

  ecvq_prep<<<CB, 256, 0, stream>>>(codebook, logits, colc);

  const int totalWaves = (b / 16) * CB;           // 12288 for b=32768
  const int blocks = (totalWaves + 7) / 8;        // 8 waves per block
  ecvq_main<<<blocks, 256, 0, stream>>>(x, codebook, colc,
                                        xhat, idxo, prob, totalWaves);

  ecvq_bits<<<1, 256, 0, stream>>>(prob, bits, b * CB);
}